// GNN_60026462929366
// MI455X (gfx1250) — compile-verified
//
#include <hip/hip_runtime.h>
#include <math.h>

typedef float v2f __attribute__((ext_vector_type(2)));
typedef float v8f __attribute__((ext_vector_type(8)));

#ifndef N_FEAT_IN
#define N_FEAT_IN 128
#define N_FEAT_HID 64
#endif

// ---------------------------------------------------------------------------
// Degree / normalization
// ---------------------------------------------------------------------------
__global__ void k_deg_init(float* __restrict__ deg, int n) {
    int i = blockIdx.x * blockDim.x + threadIdx.x;
    if (i < n) deg[i] = 1.0f;              // self-loop weight
}

__global__ void k_deg_edges(const int* __restrict__ dst, const float* __restrict__ ew,
                            float* __restrict__ deg, int E) {
    int e = blockIdx.x * blockDim.x + threadIdx.x;
    if (e < E) atomicAdd(&deg[dst[e]], ew[e]);
}

__global__ void k_make_dinv(float* __restrict__ deg_inout, int n) {
    int i = blockIdx.x * blockDim.x + threadIdx.x;
    if (i < n) {
        float v = deg_inout[i];
        deg_inout[i] = (v > 0.0f) ? rsqrtf(v) : 0.0f;
    }
}

// ---------------------------------------------------------------------------
// Layer-1 GEMM:  h1[N,64] = x[N,128] @ W1[128,64]   (fp32 WMMA 16x16x4)
// One wave per 16-row tile; wave computes the full 16x64 output strip.
// W1 staged in LDS in a K-pair-interleaved layout so each WMMA B fragment
// {W[k][n], W[k+1][n]} is one aligned 8-byte word -> single ds_load_b64
// straight into the even-aligned WMMA operand pair (no repack movs).
//   Ws[(k>>1)*128 + n*2 + (k&1)] = W1[k*64 + n]
// ---------------------------------------------------------------------------
__global__ void __launch_bounds__(256)
k_gemm1_wmma(const float* __restrict__ x, const float* __restrict__ W1,
             float* __restrict__ h1, int n_nodes) {
    __shared__ float Ws[N_FEAT_IN * N_FEAT_HID];   // 128*64*4 = 32 KB
    for (int i = threadIdx.x; i < N_FEAT_IN * N_FEAT_HID; i += blockDim.x) {
        int k = i >> 6;           // 0..127
        int n = i & 63;           // 0..63
        Ws[(k >> 1) * 128 + n * 2 + (k & 1)] = W1[i];
    }
    __syncthreads();

    const int wave = threadIdx.x >> 5;                  // 0..7
    const int lane = threadIdx.x & 31;
    const int m0   = (blockIdx.x * 8 + wave) * 16;      // tile base row
    if (m0 >= n_nodes) return;

    // A layout (16x4 f32): lanes 0-15 -> K = {k0, k0+1}; lanes 16-31 -> K = {k0+2, k0+3}
    const int mlane = lane & 15;
    const int kh    = (lane >> 4);                      // 0 or 1 (K half-pair select)
    const float* __restrict__ xrow = x + (size_t)(m0 + mlane) * N_FEAT_IN + 2 * kh;

    v8f acc0 = {}, acc1 = {}, acc2 = {}, acc3 = {};

    #pragma unroll 4
    for (int k0 = 0; k0 < N_FEAT_IN; k0 += 4) {
        // one global_load_b64: contiguous K pair
        const v2f a = *(const v2f*)(xrow + k0);

        // pair index for this lane's K half = k0/2 + kh
        const float* __restrict__ wp = &Ws[(size_t)((k0 >> 1) + kh) * 128 + mlane * 2];
        const v2f b0 = *(const v2f*)(wp + 0);    // cols  0..15
        const v2f b1 = *(const v2f*)(wp + 32);   // cols 16..31
        const v2f b2 = *(const v2f*)(wp + 64);   // cols 32..47
        const v2f b3 = *(const v2f*)(wp + 96);   // cols 48..63

        acc0 = __builtin_amdgcn_wmma_f32_16x16x4_f32(false, a, false, b0, (short)0, acc0, false, false);
        acc1 = __builtin_amdgcn_wmma_f32_16x16x4_f32(false, a, false, b1, (short)0, acc1, false, false);
        acc2 = __builtin_amdgcn_wmma_f32_16x16x4_f32(false, a, false, b2, (short)0, acc2, false, false);
        acc3 = __builtin_amdgcn_wmma_f32_16x16x4_f32(false, a, false, b3, (short)0, acc3, false, false);
    }

    // C/D layout: lane&15 = N column; lanes>=16 hold rows M = 8..15
    const int ncol  = lane & 15;
    const int mbase = m0 + (lane >> 4) * 8;
    #pragma unroll
    for (int r = 0; r < 8; ++r) {
        float* __restrict__ orow = h1 + (size_t)(mbase + r) * N_FEAT_HID + ncol;
        orow[0]  = acc0[r];
        orow[16] = acc1[r];
        orow[32] = acc2[r];
        orow[48] = acc3[r];
    }
}

// ---------------------------------------------------------------------------
// a1[n][j] = h1[n][j]*dinv[n]^2 + b1[j]   (self-loop term + bias)
// ---------------------------------------------------------------------------
__global__ void k_init_a1(const float* __restrict__ h1, const float* __restrict__ dinv,
                          const float* __restrict__ b1, float* __restrict__ a1, int total) {
    int idx = blockIdx.x * blockDim.x + threadIdx.x;
    if (idx < total) {
        int node = idx >> 6;
        int j    = idx & 63;
        float di = dinv[node];
        a1[idx] = h1[idx] * di * di + b1[j];
    }
}

// ---------------------------------------------------------------------------
// Layer-1 edge scatter: a1[dst] += h1[src] * (dinv[src]*ew*dinv[dst])
// 16 threads per edge, float4 gather + 4 float atomics each.
// ---------------------------------------------------------------------------
__global__ void k_scatter1(const int* __restrict__ src, const int* __restrict__ dst,
                           const float* __restrict__ ew, const float* __restrict__ dinv,
                           const float* __restrict__ h1, float* __restrict__ a1, int E) {
    int gid = blockIdx.x * blockDim.x + threadIdx.x;
    int e = gid >> 4;
    if (e >= E) return;
    int j4 = (gid & 15) * 4;

    int s = src[e], d = dst[e];
    float coef = dinv[s] * ew[e] * dinv[d];

    const float4 hv = *(const float4*)(h1 + (size_t)s * N_FEAT_HID + j4);
    float* out = a1 + (size_t)d * N_FEAT_HID + j4;
    atomicAdd(out + 0, hv.x * coef);
    atomicAdd(out + 1, hv.y * coef);
    atomicAdd(out + 2, hv.z * coef);
    atomicAdd(out + 3, hv.w * coef);
}

// ---------------------------------------------------------------------------
// Layer-2 fused ReLU + GEMV (64 -> 1), one wave32 per node.
// h2[n] = sum_j relu(a1[n][j]) * W2[j]
// acc2[n] = h2[n]*dinv[n]^2 + b2          (self-loop + bias init)
// ---------------------------------------------------------------------------
__global__ void __launch_bounds__(256)
k_layer2_gemv(const float* __restrict__ a1, const float* __restrict__ W2,
              const float* __restrict__ dinv, const float* __restrict__ b2,
              float* __restrict__ h2, float* __restrict__ acc2, int n) {
    int node = (blockIdx.x * blockDim.x + threadIdx.x) >> 5;
    int lane = threadIdx.x & 31;
    if (node >= n) return;

    const float* row = a1 + (size_t)node * N_FEAT_HID;
    float v = fmaxf(row[lane],      0.0f) * W2[lane] +
              fmaxf(row[lane + 32], 0.0f) * W2[lane + 32];
    #pragma unroll
    for (int off = 16; off > 0; off >>= 1)
        v += __shfl_xor(v, off, 32);

    if (lane == 0) {
        h2[node] = v;
        float di = dinv[node];
        acc2[node] = v * di * di + b2[0];
    }
}

// ---------------------------------------------------------------------------
// Layer-2 edge scatter (scalar): acc2[dst] += h2[src]*dinv[src]*ew*dinv[dst]
// ---------------------------------------------------------------------------
__global__ void k_scatter2(const int* __restrict__ src, const int* __restrict__ dst,
                           const float* __restrict__ ew, const float* __restrict__ dinv,
                           const float* __restrict__ h2, float* __restrict__ acc2, int E) {
    int e = blockIdx.x * blockDim.x + threadIdx.x;
    if (e < E) {
        int s = src[e], d = dst[e];
        atomicAdd(&acc2[d], h2[s] * dinv[s] * ew[e] * dinv[d]);
    }
}

__global__ void k_sigmoid(const float* __restrict__ acc2, float* __restrict__ out, int n) {
    int i = blockIdx.x * blockDim.x + threadIdx.x;
    if (i < n) out[i] = 1.0f / (1.0f + __expf(-acc2[i]));
}

// ---------------------------------------------------------------------------
// Host launcher
// ---------------------------------------------------------------------------
extern "C" void kernel_launch(void* const* d_in, const int* in_sizes, int n_in,
                              void* d_out, int out_size, void* d_ws, size_t ws_size,
                              hipStream_t stream) {
    const float* x   = (const float*)d_in[0];      // [N,128]
    const int*   ei  = (const int*)  d_in[1];      // [2,E]
    const float* ew  = (const float*)d_in[2];      // [E]
    const float* W1  = (const float*)d_in[3];      // [128,64]
    const float* b1  = (const float*)d_in[4];      // [64]
    const float* W2  = (const float*)d_in[5];      // [64,1]
    const float* b2  = (const float*)d_in[6];      // [1]
    float* out = (float*)d_out;                    // [N]

    const int N = in_sizes[0] / N_FEAT_IN;
    const int E = in_sizes[2];
    const int* srcIdx = ei;
    const int* dstIdx = ei + E;

    // Workspace layout (floats): dinv[N] | h1[64N] | a1[64N] | h2[N] | acc2[N]
    float* dinv = (float*)d_ws;
    float* h1   = dinv + (size_t)N;
    float* a1   = h1   + (size_t)N * N_FEAT_HID;
    float* h2   = a1   + (size_t)N * N_FEAT_HID;
    float* acc2 = h2   + (size_t)N;

    const int B = 256;
    // 1. degree + inverse-sqrt normalization (shared by both layers)
    k_deg_init <<<(N + B - 1) / B, B, 0, stream>>>(dinv, N);
    k_deg_edges<<<(E + B - 1) / B, B, 0, stream>>>(dstIdx, ew, dinv, E);
    k_make_dinv<<<(N + B - 1) / B, B, 0, stream>>>(dinv, N);

    // 2. h1 = x @ W1 via fp32 WMMA (one wave per 16-row tile, 8 waves/block)
    {
        int tiles = N / 16 + ((N % 16) ? 1 : 0);   // N=50000 -> exactly 3125
        int blocks = (tiles + 7) / 8;
        k_gemm1_wmma<<<blocks, 256, 0, stream>>>(x, W1, h1, N);
    }

    // 3. layer-1 aggregate: self-loop+bias init, then edge scatter-add
    k_init_a1<<<((size_t)N * N_FEAT_HID + B - 1) / B, B, 0, stream>>>(h1, dinv, b1, a1, N * N_FEAT_HID);
    {
        size_t threads = (size_t)E * 16;
        k_scatter1<<<(threads + B - 1) / B, B, 0, stream>>>(srcIdx, dstIdx, ew, dinv, h1, a1, E);
    }

    // 4. layer-2: fused ReLU + GEMV, self-loop init, edge scatter, sigmoid
    {
        size_t threads = (size_t)N * 32;
        k_layer2_gemv<<<(threads + 255) / 256, 256, 0, stream>>>(a1, W2, dinv, b2, h2, acc2, N);
    }
    k_scatter2<<<(E + B - 1) / B, B, 0, stream>>>(srcIdx, dstIdx, ew, dinv, h2, acc2, E);
    k_sigmoid <<<(N + B - 1) / B, B, 0, stream>>>(acc2, out, N);

    (void)n_in; (void)out_size; (void)ws_size;
}